// Z_73813307949717
// MI455X (gfx1250) — compile-verified
//
#include <hip/hip_runtime.h>
#include <stdint.h>

// ---------------------------------------------------------------------------
// Qudit Z-gate diagonal phase multiply (DIM=2, qudits {0,5,10} of L=14).
//   out[0] = c*xr - s*xi ; out[1] = c*xi + s*xr
// (c,s) depend only on tot = popcount(row & 0x2108) in {0,1,2,3}, and row is
// wave-uniform -> phase computed entirely on the SALU via readfirstlane.
// Pure streaming problem: 512 MB traffic, ~22 us at 23.3 TB/s.
// Data path: gfx1250 async global->LDS (ASYNCcnt, NT hint) + ds_load_b128
// readback + non-temporal B128 stores.
// ---------------------------------------------------------------------------

typedef float v4f __attribute__((ext_vector_type(4)));

#define NROWS        16384              // N = 2^14
#define BCOLS        2048               // batch
#define UNITS_ROW    (BCOLS / 4)        // 512 float4 per row
#define ROW_SHIFT    9                  // log2(512)
#define PLANE_UNITS  (NROWS * UNITS_ROW)    // 8,388,608 float4 per plane
#define PHASE_MASK   0x2108u            // bits 13, 8, 3  (qudits 0,5,10)
#define BLOCK        256

__global__ __launch_bounds__(BLOCK) void zphase_kernel(
    const v4f* __restrict__ xr,
    const v4f* __restrict__ xi,
    v4f* __restrict__ out)
{
    __shared__ v4f smem_r[BLOCK];
    __shared__ v4f smem_i[BLOCK];

    const uint32_t t   = threadIdx.x;
    const uint32_t idx = blockIdx.x * BLOCK + t;   // float4 unit index, exact cover

    // LDS byte offsets (low 32 bits of generic pointer == addrspace(3) offset)
    uint32_t lds_r = (uint32_t)(uintptr_t)&smem_r[t];
    uint32_t lds_i = (uint32_t)(uintptr_t)&smem_i[t];
    // Per-lane 64-bit global addresses; consecutive lanes -> contiguous 512B/wave
    uint64_t ga_r  = (uint64_t)(uintptr_t)(xr + idx);
    uint64_t ga_i  = (uint64_t)(uintptr_t)(xi + idx);

    // gfx1250 async copy global -> LDS, tracked by ASYNCcnt (16B per lane).
    // Read-once input: non-temporal hint, don't displace L2.
    asm volatile("global_load_async_to_lds_b128 %0, %1, off th:TH_LOAD_NT"
                 :: "v"(lds_r), "v"(ga_r) : "memory");
    asm volatile("global_load_async_to_lds_b128 %0, %1, off th:TH_LOAD_NT"
                 :: "v"(lds_i), "v"(ga_i) : "memory");
    asm volatile("s_wait_asynccnt 0" ::: "memory");

    // Each lane reads back only its own slot: no cross-wave sharing, no barrier.
    v4f vr = smem_r[t];
    v4f vi = smem_i[t];

    // row = idx >> 9 is wave-uniform (512-unit rows are 32-lane aligned):
    // compute the phase on the scalar unit, selects become s_cselect.
    uint32_t srow = (uint32_t)__builtin_amdgcn_readfirstlane((int)(idx >> ROW_SHIFT));
    uint32_t tot  = (uint32_t)__popc(srow & PHASE_MASK);

    // Exact float32 values of cos/sin(float32(pi) * tot), matching reference
    float c = (tot & 1u) ? -1.0f : 1.0f;
    float s = (tot & 2u) ? ((tot & 1u) ? -2.3849761e-8f : 1.7484555e-7f)
                         : ((tot & 1u) ? -8.7422777e-8f : 0.0f);
    float ns = -s;

    v4f o_r, o_i;
    o_r.x = fmaf(c, vr.x, ns * vi.x);
    o_r.y = fmaf(c, vr.y, ns * vi.y);
    o_r.z = fmaf(c, vr.z, ns * vi.z);
    o_r.w = fmaf(c, vr.w, ns * vi.w);
    o_i.x = fmaf(c, vi.x,  s * vr.x);
    o_i.y = fmaf(c, vi.y,  s * vr.y);
    o_i.z = fmaf(c, vi.z,  s * vr.z);
    o_i.w = fmaf(c, vi.w,  s * vr.w);

    // Write-once output: non-temporal B128 stores
    __builtin_nontemporal_store(o_r, &out[idx]);
    __builtin_nontemporal_store(o_i, &out[PLANE_UNITS + idx]);
}

extern "C" void kernel_launch(void* const* d_in, const int* in_sizes, int n_in,
                              void* d_out, int out_size, void* d_ws, size_t ws_size,
                              hipStream_t stream) {
    (void)in_sizes; (void)n_in; (void)out_size; (void)d_ws; (void)ws_size;
    const v4f* xr = (const v4f*)d_in[0];
    const v4f* xi = (const v4f*)d_in[1];
    v4f* out      = (v4f*)d_out;

    const int grid = PLANE_UNITS / BLOCK;   // 32768 blocks, exact cover
    zphase_kernel<<<grid, BLOCK, 0, stream>>>(xr, xi, out);
}